// SpectralGatingNetwork_26371099198263
// MI455X (gfx1250) — compile-verified
//
#include <hip/hip_runtime.h>

// ---------------------------------------------------------------------------
// SpectralGatingNetwork (Haar DWT -> gate LL -> iDWT), MI455X / gfx1250.
//
// Algebraic collapse: out = x + (w-1)/4 * (2x2 quad sum) on rows [32,192);
// rows [0,32) are a straight copy. ~302 MB of HBM traffic -> ~13 us at
// 23.3 TB/s -> memory-bound elementwise stream. No matmul -> no WMMA.
// CDNA5-specific paths: global_load_async_to_lds_b128 (weight slab staging
// into LDS) + s_wait_asynccnt, non-temporal b128 loads/stores for the
// streamed tensor. Division-free addressing keeps VALU issue pressure low.
// ---------------------------------------------------------------------------

typedef float __attribute__((ext_vector_type(4))) v4f;
typedef int   __attribute__((ext_vector_type(4))) v4i;

#define AS1 __attribute__((address_space(1)))
#define AS3 __attribute__((address_space(3)))

#ifndef __has_builtin
#define __has_builtin(x) 0
#endif

#if __has_builtin(__builtin_amdgcn_global_load_async_to_lds_b128) && \
    __has_builtin(__builtin_amdgcn_s_wait_asynccnt)
#define SGN_ASYNC_LDS 1
#else
#define SGN_ASYNC_LDS 0
#endif

constexpr int kB        = 256;
constexpr int kRows     = 192;   // lens_x + H*W = 32 + 160
constexpr int kC        = 768;
constexpr int kLens     = 32;
constexpr int kH        = 16;
constexpr int kW        = 10;
constexpr int kQPR      = kC / 4;          // float4 quads per row = 192
constexpr int kJ        = kW / 2;          // column pairs = 5
constexpr int kQPB      = kJ * kQPR;       // quads per (b,h) block = 960
constexpr int kThreads  = kQPR;            // 192 threads = 6 waves (wave32)

// ---- passthrough copy of rows [0, 32) -------------------------------------
// grid = (24, 256): blockIdx.y = batch, 24*256 threads cover 6144 prefix quads
__global__ __launch_bounds__(256) void sgn_copy_prefix(const v4f* __restrict__ x,
                                                       v4f* __restrict__ out) {
    const size_t off = (size_t)blockIdx.y * (kRows * kQPR)     // batch stride 36864
                     + (size_t)blockIdx.x * 256 + threadIdx.x; // [0, 6144) in-batch
    v4f v = __builtin_nontemporal_load(x + off);
    __builtin_nontemporal_store(v, out + off);
}

// ---- gated Haar transform on rows [32, 192) --------------------------------
__global__ __launch_bounds__(kThreads) void sgn_transform(const v4f* __restrict__ x,
                                                          const v4f* __restrict__ cw,
                                                          v4f* __restrict__ out) {
    __shared__ v4f wlds[kQPB];                     // 15 KB: weight slab for this h

    const int tid = threadIdx.x;                   // float4 lane along C: m = tid
    const int b   = blockIdx.x >> 4;               // grid = B * 16
    const int h   = blockIdx.x & 15;
    const int wbase = h * kQPB;                    // float4 index into complex_weight

#if SGN_ASYNC_LDS
    // Stage (h)-slab of complex_weight into LDS via async global->LDS DMA path.
#pragma unroll
    for (int s = 0; s < kJ; ++s) {                 // 960 / 192 = 5 transfers/thread
        const int i = s * kThreads + tid;
        const AS1 v4f* gsrc = (const AS1 v4f*)(cw + wbase + i);
        AS3 v4f*       ldst = (AS3 v4f*)(&wlds[i]);
        __builtin_amdgcn_global_load_async_to_lds_b128(
            (AS1 v4i*)gsrc, (AS3 v4i*)ldst, /*offset=*/0, /*cpol=*/0);
    }
    __builtin_amdgcn_s_wait_asynccnt(0);
    __syncthreads();
#else
#pragma unroll
    for (int s = 0; s < kJ; ++s) {
        const int i = s * kThreads + tid;
        wlds[i] = cw[wbase + i];
    }
    __syncthreads();
#endif

    // float4 offset of row (b, 32 + h*10), plus this thread's channel quad
    const size_t baseq = (size_t)(b * kRows + kLens + h * kW) * kQPR + tid;

#pragma unroll
    for (int j = 0; j < kJ; ++j) {                 // column pair index (unrolled)
        const size_t oe = baseq + (size_t)(2 * j) * kQPR;      // even column row
        const size_t oo = oe + kQPR;                           // odd column row

        const v4f e = __builtin_nontemporal_load(x + oe);
        const v4f o = __builtin_nontemporal_load(x + oo);

        // wlds[j*192+tid] = [re(k0), im(k0), re(k0+1), im(k0+1)] for k0 = 2*tid
        const v4f wv = wlds[j * kThreads + tid];
        const float g0 = (wv.x - 1.0f) * 0.25f;
        const float g1 = (wv.z - 1.0f) * 0.25f;

        const float d0 = g0 * ((e.x + e.y) + (o.x + o.y));
        const float d1 = g1 * ((e.z + e.w) + (o.z + o.w));

        v4f re;  re.x = e.x + d0; re.y = e.y + d0; re.z = e.z + d1; re.w = e.w + d1;
        v4f ro;  ro.x = o.x + d0; ro.y = o.y + d0; ro.z = o.z + d1; ro.w = o.w + d1;

        __builtin_nontemporal_store(re, out + oe);
        __builtin_nontemporal_store(ro, out + oo);
    }
}

extern "C" void kernel_launch(void* const* d_in, const int* in_sizes, int n_in,
                              void* d_out, int out_size, void* d_ws, size_t ws_size,
                              hipStream_t stream) {
    (void)in_sizes; (void)n_in; (void)out_size; (void)d_ws; (void)ws_size;
    const v4f* x  = (const v4f*)d_in[0];   // (256, 192, 768) fp32
    const v4f* cw = (const v4f*)d_in[1];   // (16, 5, 384, 2) fp32
    v4f* out = (v4f*)d_out;                // (256, 192, 768) fp32

    // prefix copy: rows [0,32) of each batch; grid (24, 256) x 256 threads
    dim3 cgrid(kLens * kQPR / 256, kB);
    sgn_copy_prefix<<<cgrid, 256, 0, stream>>>(x, out);

    // transform: one block per (b, h)
    sgn_transform<<<kB * kH, kThreads, 0, stream>>>(x, cw, out);
}